// DHPC_11141145166136
// MI455X (gfx1250) — compile-verified
//
#include <hip/hip_runtime.h>
#include <math.h>

typedef float v2f __attribute__((ext_vector_type(2)));
typedef float v8f __attribute__((ext_vector_type(8)));

#define N0 4096
#define N1 4096
#define N2 2048
#define N3 512
#define NSTEPS 50
#define CHUNKS 16

// workspace layout (float offsets)
#define OFF_RA1 0
#define OFF_RO1 (OFF_RA1 + N1)
#define OFF_RA2 (OFF_RO1 + N1)
#define OFF_RO2 (OFF_RA2 + N2)
#define OFF_RAL (OFF_RO2 + N2)
#define OFF_ROL (OFF_RAL + N3)
#define OFF_E0  (OFF_ROL + N3)
#define OFF_E1  (OFF_E0 + N0)
#define OFF_E2  (OFF_E1 + N1)
#define OFF_P1  (OFF_E2 + N2)            // CHUNKS*N1
#define OFF_P2  (OFF_P1 + CHUNKS * N1)   // CHUNKS*N2
#define OFF_PL  (OFF_P2 + CHUNKS * N2)   // CHUNKS*N3
// total = 130048 floats = ~508 KB of ws

// ---------------------------------------------------------------------------
// init: ra=-2, ro=tanh(-2); raL = cat_mem*0.1, roL = tanh(-2)
// ---------------------------------------------------------------------------
__global__ __launch_bounds__(256) void pc_init_kernel(const float* __restrict__ cat_mem,
                                                      float* __restrict__ ws) {
  int gid = blockIdx.x * 256 + threadIdx.x;
  const float T0 = tanhf(-2.0f);
  if (gid < N1) {
    ws[OFF_RA1 + gid] = -2.0f;
    ws[OFF_RO1 + gid] = T0;
  } else if (gid < N1 + N2) {
    int j = gid - N1;
    ws[OFF_RA2 + j] = -2.0f;
    ws[OFF_RO2 + j] = T0;
  } else if (gid < N1 + N2 + N3) {
    int j = gid - N1 - N2;
    ws[OFF_RAL + j] = cat_mem[j] * 0.1f;
    ws[OFF_ROL + j] = T0;
  }
}

// ---------------------------------------------------------------------------
// Forward mat-vecs via V_WMMA_F32_16X16X4_F32.
// tile t = 16 output rows of one of the 3 products (256 + 256 + 128 = 640 tiles).
// 4 waves per tile, each owning a quarter of K; 2 tiles per 256-thread block.
// Vector sits in B column 0 (lanes 0/16), other columns zeroed by a VALU mask
// so EXEC remains all-ones across every WMMA (ISA §7.12 requirement).
// 4 independent accumulator chains (K step 16) to hide WMMA RAW latency at
// low occupancy; each WMMA consumes a 256B weight tile -> L2-bandwidth fed.
// Epilogue: e = sub - W@v fused.
// ---------------------------------------------------------------------------
__global__ __launch_bounds__(256) void pc_fwd_kernel(const float* __restrict__ frame,
                                                     const float* __restrict__ W0,
                                                     const float* __restrict__ W1,
                                                     const float* __restrict__ W2,
                                                     float* __restrict__ ws) {
  __shared__ float red[2][4][16];
  const int tid = threadIdx.x;
  const int lane = tid & 31;
  const int wib = tid >> 5;     // wave in block: 0..7
  const int tib = wib >> 2;     // tile in block: 0..1
  const int ksl = wib & 3;      // K-slice: 0..3
  const int half = lane >> 4;   // 0: K pair {0,1}, 1: K pair {2,3}
  const int r = lane & 15;      // row within 16-row tile / N index

  const int tile = blockIdx.x * 2 + tib;   // 0..639

  const float* W;
  const float* v;
  int m, row0;
  if (tile < 256) {                // t0 = W0 @ ro1
    W = W0; v = ws + OFF_RO1; m = N1; row0 = tile * 16;
  } else if (tile < 512) {         // t1 = W1 @ ro2
    W = W1; v = ws + OFF_RO2; m = N2; row0 = (tile - 256) * 16;
  } else {                         // t2 = W2 @ roL
    W = W2; v = ws + OFF_ROL; m = N3; row0 = (tile - 512) * 16;
  }

  const int klen = m >> 2;         // quarter of K per wave (>=128, mult of 16)
  const int kbeg = ksl * klen;
  const int kend = kbeg + klen;

  // A tile: lane r holds row (row0+r), K pair selected by `half`
  const float* arow = W + (size_t)(row0 + r) * m + 2 * half;
  // B column 0: lanes 0 (K0,K1) and 16 (K2,K3) carry the vector, rest zero
  const float* vrow = v + 2 * half;
  const float msk = (r == 0) ? 1.0f : 0.0f;

  v8f acc0 = {};
  v8f acc1 = {};
  v8f acc2 = {};
  v8f acc3 = {};
  for (int k0 = kbeg; k0 < kend; k0 += 16) {
    v2f a0 = *(const v2f*)(arow + k0);
    v2f b0 = *(const v2f*)(vrow + k0) * msk;
    v2f a1 = *(const v2f*)(arow + k0 + 4);
    v2f b1 = *(const v2f*)(vrow + k0 + 4) * msk;
    v2f a2 = *(const v2f*)(arow + k0 + 8);
    v2f b2 = *(const v2f*)(vrow + k0 + 8) * msk;
    v2f a3 = *(const v2f*)(arow + k0 + 12);
    v2f b3 = *(const v2f*)(vrow + k0 + 12) * msk;
    acc0 = __builtin_amdgcn_wmma_f32_16x16x4_f32(false, a0, false, b0,
                                                 (short)0, acc0, false, false);
    acc1 = __builtin_amdgcn_wmma_f32_16x16x4_f32(false, a1, false, b1,
                                                 (short)0, acc1, false, false);
    acc2 = __builtin_amdgcn_wmma_f32_16x16x4_f32(false, a2, false, b2,
                                                 (short)0, acc2, false, false);
    acc3 = __builtin_amdgcn_wmma_f32_16x16x4_f32(false, a3, false, b3,
                                                 (short)0, acc3, false, false);
  }

  // D column 0: rows 0..7 in lane 0 VGPR 0..7; rows 8..15 in lane 16 VGPR 0..7
  if (r == 0) {
#pragma unroll
    for (int q = 0; q < 8; ++q)
      red[tib][ksl][8 * half + q] = (acc0[q] + acc1[q]) + (acc2[q] + acc3[q]);
  }
  __syncthreads();

  // reduce the 4 K-slices and fuse the error epilogue
  if (tid < 32) {
    const int tsel = tid >> 4;
    const int rr = tid & 15;
    const int t2 = blockIdx.x * 2 + tsel;
    const float* sub;
    float* e;
    int rb;
    if (t2 < 256) {
      sub = frame;        e = ws + OFF_E0; rb = t2 * 16;
    } else if (t2 < 512) {
      sub = ws + OFF_RO1; e = ws + OFF_E1; rb = (t2 - 256) * 16;
    } else {
      sub = ws + OFF_RO2; e = ws + OFF_E2; rb = (t2 - 512) * 16;
    }
    float s = red[tsel][0][rr] + red[tsel][1][rr] + red[tsel][2][rr] + red[tsel][3][rr];
    const int row = rb + rr;
    e[row] = sub[row] - s;
  }
}

// ---------------------------------------------------------------------------
// Transposed mat-vecs: bu = W^T @ e. Thread-per-column, streaming rows
// coalesced (consecutive lanes -> consecutive columns). Row dim split into
// 16 chunks -> deterministic partial sums (no float atomics).
// 16*4096 + 16*2048 + 16*512 = 106496 threads.
// ---------------------------------------------------------------------------
__global__ __launch_bounds__(256) void pc_bwd_kernel(const float* __restrict__ W0,
                                                     const float* __restrict__ W1,
                                                     const float* __restrict__ W2,
                                                     float* __restrict__ ws) {
  const int gid = blockIdx.x * 256 + threadIdx.x;
  const float* W;
  const float* e;
  float* p;
  int m, j, c, rpc;
  if (gid < CHUNKS * N1) {                       // bu1 = W0^T e0
    c = gid >> 12; j = gid & (N1 - 1);
    W = W0; e = ws + OFF_E0; p = ws + OFF_P1; m = N1; rpc = N0 / CHUNKS;   // 256
  } else if (gid < CHUNKS * (N1 + N2)) {         // bu2 = W1^T e1
    int t = gid - CHUNKS * N1;
    c = t >> 11; j = t & (N2 - 1);
    W = W1; e = ws + OFF_E1; p = ws + OFF_P2; m = N2; rpc = N1 / CHUNKS;   // 256
  } else {                                       // buL = W2^T e2
    int t = gid - CHUNKS * (N1 + N2);
    c = t >> 9; j = t & (N3 - 1);
    W = W2; e = ws + OFF_E2; p = ws + OFF_PL; m = N3; rpc = N2 / CHUNKS;   // 128
  }
  const int i0 = c * rpc;
  const float* wp = W + (size_t)i0 * m + j;
  const float* ep = e + i0;
  float s = 0.0f;
#pragma unroll 4
  for (int i = 0; i < rpc; ++i)
    s = fmaf(wp[(size_t)i * m], ep[i], s);
  p[c * m + j] = s;
}

// ---------------------------------------------------------------------------
// Elementwise state update: reduce 16 partials, integrate, tanh.
// Output layer also applies decay_w * cat_mem (decay from device-resident
// inference_steps) and streams roL to d_out every step (last step wins).
// ---------------------------------------------------------------------------
__global__ __launch_bounds__(256) void pc_upd_kernel(const float* __restrict__ cat_mem,
                                                     const int* __restrict__ steps_p,
                                                     float* __restrict__ ws,
                                                     float* __restrict__ out) {
  const int gid = blockIdx.x * 256 + threadIdx.x;
  if (gid < N1) {
    const float* p = ws + OFF_P1;
    float bu = 0.0f;
#pragma unroll
    for (int c = 0; c < CHUNKS; ++c) bu += p[c * N1 + gid];
    float ra = ws[OFF_RA1 + gid] + 0.05f * (bu - ws[OFF_E1 + gid]);
    ws[OFF_RA1 + gid] = ra;
    ws[OFF_RO1 + gid] = tanhf(ra);
  } else if (gid < N1 + N2) {
    const int j = gid - N1;
    const float* p = ws + OFF_P2;
    float bu = 0.0f;
#pragma unroll
    for (int c = 0; c < CHUNKS; ++c) bu += p[c * N2 + j];
    float ra = ws[OFF_RA2 + j] + 0.05f * (bu - ws[OFF_E2 + j]);
    ws[OFF_RA2 + j] = ra;
    ws[OFF_RO2 + j] = tanhf(ra);
  } else if (gid < N1 + N2 + N3) {
    const int j = gid - N1 - N2;
    const float* p = ws + OFF_PL;
    float bu = 0.0f;
#pragma unroll
    for (int c = 0; c < CHUNKS; ++c) bu += p[c * N3 + j];
    const float decay = 0.1f * expf(-(float)(*steps_p) * (1.0f / 30.0f));
    float ra = ws[OFF_RAL + j] + 0.05f * bu + decay * cat_mem[j];
    ws[OFF_RAL + j] = ra;
    float ro = tanhf(ra);
    ws[OFF_ROL + j] = ro;
    out[j] = ro;
  }
}

// ---------------------------------------------------------------------------
extern "C" void kernel_launch(void* const* d_in, const int* in_sizes, int n_in,
                              void* d_out, int out_size, void* d_ws, size_t ws_size,
                              hipStream_t stream) {
  const float* frame   = (const float*)d_in[0];
  const float* cat_mem = (const float*)d_in[1];
  const float* W0      = (const float*)d_in[2];
  const float* W1      = (const float*)d_in[3];
  const float* W2      = (const float*)d_in[4];
  const int*   steps   = (const int*)d_in[5];
  float* ws  = (float*)d_ws;
  float* out = (float*)d_out;
  (void)in_sizes; (void)n_in; (void)out_size; (void)ws_size;

  pc_init_kernel<<<26, 256, 0, stream>>>(cat_mem, ws);
  for (int s = 0; s < NSTEPS; ++s) {
    pc_fwd_kernel<<<320, 256, 0, stream>>>(frame, W0, W1, W2, ws);  // 640 tiles * 4 waves
    pc_bwd_kernel<<<416, 256, 0, stream>>>(W0, W1, W2, ws);         // 106496 threads
    pc_upd_kernel<<<26, 256, 0, stream>>>(cat_mem, steps, ws, out); // 6656 threads
  }
}